// MultiLinear_45157286150938
// MI455X (gfx1250) — compile-verified
//
#include <hip/hip_runtime.h>
#include <hip/hip_bf16.h>

// MultiLinear: out[c,b,d] = sum_a input[a,b] * W[rc[c],a,d] + bias[rc[c],d]
// C=128, B=1024, A(K)=512, D(N)=512, fp32.
// Roofline: 68.7 GFLOP vs ~386 MB HBM (gathered W 128MB + out 256MB + X 2MB)
// -> 16.6us floor @ 23.3 TB/s, only ~4.1 TF/s compute needed. Memory-bound:
// keep fp32 precision, V_WMMA_F32_16X16X4_F32 for the math, and pipeline the
// HBM stream (double-buffered LDS, loads issued a full slab ahead).

typedef __attribute__((ext_vector_type(2))) float v2f;
typedef __attribute__((ext_vector_type(4))) float v4f;
typedef __attribute__((ext_vector_type(8))) float v8f;

constexpr int IN_F   = 512;   // A / K
constexpr int OUT_F  = 512;   // D / N
constexpr int BATCH  = 1024;  // B / M
constexpr int KT     = 32;    // K slab per LDS stage
constexpr int NSLAB  = IN_F / KT;  // 16
constexpr int ROWP   = 144;   // LDS row pitch (floats): 2*ROWP % 64 == 32 -> half-wave
                              // K-offset (+2 rows) lands 32 banks away, conflict-free
constexpr int BM     = 128;   // block tile M
constexpr int BN     = 128;   // block tile N

__global__ __launch_bounds__(256)
void MultiLinear_45157286150938_kernel(const float* __restrict__ input,
                                       const int*   __restrict__ regions,
                                       const float* __restrict__ wt,
                                       const float* __restrict__ bias,
                                       float*       __restrict__ out)
{
    __shared__ float Xs[2][KT * ROWP];   // Xs[buf][k][m], k-major, padded rows
    __shared__ float Ws[2][KT * ROWP];   // Ws[buf][k][n]

    const int c     = blockIdx.y;          // region 0..127
    const int t     = blockIdx.x;          // 0..31
    const int mBase = (t >> 2) * BM;       // 8 M-tiles over BATCH=1024
    const int nBase = (t & 3) * BN;        // 4 N-tiles over OUT_F=512
    const int rc    = regions[c];

    const int tid    = threadIdx.x;
    const int lane   = tid & 31;
    const int wv     = tid >> 5;           // wave 0..7
    const int laneM  = lane & 15;          // row/col within 16
    const int koff   = (lane >> 4) << 1;   // half-wave K offset: 0 or 2
    const int rowOff = (lane >> 4) << 3;   // C/D row offset: 0 or 8

    const int mw = (wv >> 2) * 64;         // wave patch: 64 (M) x 32 (N)
    const int nw = (wv & 3) * 32;

    const float* wbase = wt + (size_t)rc * (IN_F * OUT_F);

    // Per-thread staging coordinates: thread copies rows kr = wv + 8i, cols cc..cc+3.
    const int cc = (tid & 31) << 2;        // 0..124 step 4

    v8f acc[4][2];
    #pragma unroll
    for (int mi = 0; mi < 4; ++mi)
        #pragma unroll
        for (int ni = 0; ni < 2; ++ni)
            acc[mi][ni] = (v8f){0.f, 0.f, 0.f, 0.f, 0.f, 0.f, 0.f, 0.f};

    // ---- prologue: stage slab 0 into buffer 0 ----
    #pragma unroll
    for (int i = 0; i < 4; ++i) {
        const int kr = wv + 8 * i;         // 0..31
        v4f xv = *(const v4f*)(input + (size_t)kr * BATCH + mBase + cc);
        v4f wv4 = *(const v4f*)(wbase + (size_t)kr * OUT_F + nBase + cc);
        *(v4f*)(&Xs[0][kr * ROWP + cc]) = xv;
        *(v4f*)(&Ws[0][kr * ROWP + cc]) = wv4;
    }
    __syncthreads();

    for (int s = 0; s < NSLAB; ++s) {
        const int cur  = s & 1;
        const bool more = (s + 1 < NSLAB);

        // Issue next slab's global loads before the compute so HBM latency
        // overlaps the 64 WMMAs + LDS reads of the current slab.
        v4f xr[4], wr[4];
        if (more) {
            const int k1 = (s + 1) * KT;
            #pragma unroll
            for (int i = 0; i < 4; ++i) {
                const int kr = wv + 8 * i;
                xr[i] = *(const v4f*)(input + (size_t)(k1 + kr) * BATCH + mBase + cc);
                wr[i] = *(const v4f*)(wbase + (size_t)(k1 + kr) * OUT_F + nBase + cc);
            }
        }

        const float* xb = &Xs[cur][0];
        const float* wb = &Ws[cur][0];

        #pragma unroll
        for (int k4 = 0; k4 < KT; k4 += 4) {
            // B fragments (4x16 K-major): VGPR0 = K {k4+koff}, VGPR1 = K {k4+koff+1}
            v2f bfrag[2];
            #pragma unroll
            for (int ni = 0; ni < 2; ++ni) {
                const int col = nw + ni * 16 + laneM;
                bfrag[ni].x = wb[(k4 + koff)     * ROWP + col];
                bfrag[ni].y = wb[(k4 + koff + 1) * ROWP + col];
            }
            // A fragments (16x4): lanes 0-15 rows M, K {k,k+1}; lanes 16-31 K {k+2,k+3}
            v2f afrag[4];
            #pragma unroll
            for (int mi = 0; mi < 4; ++mi) {
                const int row = mw + mi * 16 + laneM;
                afrag[mi].x = xb[(k4 + koff)     * ROWP + row];
                afrag[mi].y = xb[(k4 + koff + 1) * ROWP + row];
            }
            #pragma unroll
            for (int mi = 0; mi < 4; ++mi)
                #pragma unroll
                for (int ni = 0; ni < 2; ++ni)
                    acc[mi][ni] = __builtin_amdgcn_wmma_f32_16x16x4_f32(
                        /*neg_a=*/false, afrag[mi],
                        /*neg_b=*/false, bfrag[ni],
                        /*c_mod=*/(short)0, acc[mi][ni],
                        /*reuse_a=*/false, /*reuse_b=*/false);
        }

        // Park the prefetched slab into the other buffer. Safe without a
        // pre-store barrier: its previous readers finished before the barrier
        // that ended iteration s-1.
        if (more) {
            const int nxt = cur ^ 1;
            #pragma unroll
            for (int i = 0; i < 4; ++i) {
                const int kr = wv + 8 * i;
                *(v4f*)(&Xs[nxt][kr * ROWP + cc]) = xr[i];
                *(v4f*)(&Ws[nxt][kr * ROWP + cc]) = wr[i];
            }
        }
        __syncthreads();
    }

    // Bias add + nontemporal store (write-once 256MB stream; keep W/X in L2).
    // C/D layout: VGPR v -> row (v + rowOff), col = laneM.
    #pragma unroll
    for (int ni = 0; ni < 2; ++ni) {
        const int ncol = nBase + nw + ni * 16 + laneM;
        const float bv = bias[rc * OUT_F + ncol];
        #pragma unroll
        for (int mi = 0; mi < 4; ++mi) {
            const size_t rowBase = (size_t)c * BATCH + mBase + mw + mi * 16 + rowOff;
            float* po = out + rowBase * OUT_F + ncol;
            #pragma unroll
            for (int v = 0; v < 8; ++v)
                __builtin_nontemporal_store(acc[mi][ni][v] + bv, po + (size_t)v * OUT_F);
        }
    }
}

extern "C" void kernel_launch(void* const* d_in, const int* in_sizes, int n_in,
                              void* d_out, int out_size, void* d_ws, size_t ws_size,
                              hipStream_t stream) {
    (void)in_sizes; (void)n_in; (void)out_size; (void)d_ws; (void)ws_size;
    const float* input   = (const float*)d_in[0];
    const int*   regions = (const int*)d_in[1];
    const float* wt      = (const float*)d_in[2];
    const float* bias    = (const float*)d_in[3];
    float*       out     = (float*)d_out;

    dim3 grid(32, 128);   // 32 tiles (8 M x 4 N) per region, 128 regions
    dim3 block(256);      // 8 wave32
    MultiLinear_45157286150938_kernel<<<grid, block, 0, stream>>>(
        input, regions, wt, bias, out);
}